// MambaExpert_41798621725046
// MI455X (gfx1250) — compile-verified
//
#include <hip/hip_runtime.h>
#include <hip/hip_bf16.h>
#include <math.h>

// ---- problem constants (from reference) ----
#define D_MODEL   1024
#define D_STATE   128
#define D_CONV    4
#define HEADDIM   64
#define D_INNER   2048
#define NHEADS    32
#define CONV_DIM  2304              // D_INNER + 2*D_STATE
#define N1        4384              // 2*D_INNER + 2*D_STATE + NHEADS
#define BATCH     2
#define SEQ       1000
#define M_ROWS    2000              // BATCH*SEQ
#define CHUNK     10
#define NCHUNK    100               // SEQ / CHUNK

typedef float v2f  __attribute__((ext_vector_type(2)));
typedef float v8f  __attribute__((ext_vector_type(8)));
typedef unsigned int u32x4 __attribute__((ext_vector_type(4)));
typedef int   i32x8 __attribute__((ext_vector_type(8)));
typedef int   i32x4 __attribute__((ext_vector_type(4)));

// LDS row stride for the TDM-staged weight tile: 64 data DWORDs + 2 pad DWORDs
#define BROW 66
#define KCHUNK 64
#define LDS_BYTES (32 * BROW * 4)   // 8448 B

// -----------------------------------------------------------------------------
// GEMM: C[m,n] = sum_k A[m,k] * Bw[n,k]   (A: MxK row-major, Bw: NxK row-major)
// fp32 WMMA 16x16x4. Wave computes a 16x32 tile (2 accumulators).
// Block = 4 waves stacked along M; all waves share the same 32 weight rows,
// which are staged into LDS by the Tensor Data Mover (one DMA per K-chunk,
// issued by wave 0). TDM padding (2 DWORDs per 64) gives LDS row stride 66
// -> conflict-free ds_load_b64 fragment reads (lm*66 mod 64 distinct).
// Fragment layout per CDNA5 ISA 7.12.2:
//   A 16x4 : lane(0-15)->M, VGPR0 = K=2*hi, VGPR1 = K=2*hi+1  (hi = lane>=16)
//   B 4x16 : lane(0-15)->N, VGPR0 = K=2*hi, VGPR1 = K=2*hi+1
//   C 16x16: VGPR r -> row (r + 8*hi), col = lane&15
// -----------------------------------------------------------------------------
__global__ __launch_bounds__(128) void gemm_f32_wmma(
    const float* __restrict__ A, const float* __restrict__ Bw,
    float* __restrict__ C, int Mtiles, int K, int N)
{
    extern __shared__ float sB[];   // [32][BROW], written by TDM

    const int wave = threadIdx.x >> 5;
    const int lane = threadIdx.x & 31;
    const int lm   = lane & 15;
    const int hi   = lane >> 4;

    int mt = blockIdx.x * 4 + wave;
    const bool active = (mt < Mtiles);
    if (!active) mt = Mtiles - 1;          // clamp: keep wave alive for barriers
    const int m0 = mt * 16;
    const int n0 = blockIdx.y * 32;

    const float* arow = A + (size_t)(m0 + lm) * K + 2 * hi;
    const unsigned long long bbase =
        (unsigned long long)(const void*)Bw + 4ull * (size_t)n0 * (size_t)K;

    v8f acc0 = {};
    v8f acc1 = {};

    for (int k0 = 0; k0 < K; k0 += KCHUNK) {
        if (wave == 0) {
            // ---- Tensor DMA: stage Bw[n0..n0+31][k0..k0+63] into LDS ----
            unsigned long long ga = bbase + 4ull * (unsigned)k0;
            u32x4 g0;
            i32x8 g1;
            i32x4 gz4 = {0, 0, 0, 0};
            i32x8 gz8 = {0, 0, 0, 0, 0, 0, 0, 0};
            g0[0] = 1u;                                   // count=1, user mode
            g0[1] = 0u;                                   // lds_addr = dyn-LDS base
            g0[2] = (unsigned)(ga & 0xffffffffu);         // global_addr[31:0]
            g0[3] = (unsigned)((ga >> 32) & 0x01ffffffu)  // global_addr[56:32]
                    | (2u << 30);                         // type = 2 (image)
            // data_size=4B (2), pad_enable, pad_interval=64 DW (5), pad_amount=2 DW (1)
            g1[0] = (int)((2u << 16) | (1u << 20) | (5u << 22) | (1u << 25));
            g1[1] = (int)(((unsigned)K & 0xffffu) << 16);           // tensor_dim0 lo
            g1[2] = (int)(((unsigned)K >> 16) |
                          (((unsigned)N & 0xffffu) << 16));         // dim0 hi | dim1 lo
            g1[3] = (int)(((unsigned)N >> 16) | (64u << 16));       // dim1 hi | tile_dim0=64
            g1[4] = 32;                                             // tile_dim1=32
            g1[5] = K;                                              // tensor_dim0_stride lo
            g1[6] = 0;
            g1[7] = 0;
            __builtin_amdgcn_tensor_load_to_lds(g0, g1, gz4, gz4, gz8, 0);
            __builtin_amdgcn_s_wait_tensorcnt(0);
        }
        __syncthreads();

        #pragma unroll 4
        for (int kk = 0; kk < KCHUNK; kk += 4) {
            v2f a  = *(const v2f*)(arow + k0 + kk);
            v2f b0 = *(const v2f*)&sB[lm * BROW + kk + 2 * hi];
            v2f b1 = *(const v2f*)&sB[(lm + 16) * BROW + kk + 2 * hi];
            acc0 = __builtin_amdgcn_wmma_f32_16x16x4_f32(
                       false, a, false, b0, (short)0, acc0, false, false);
            acc1 = __builtin_amdgcn_wmma_f32_16x16x4_f32(
                       false, a, false, b1, (short)0, acc1, false, false);
        }
        __syncthreads();
    }

    if (active) {
        float* crow = C + (size_t)(m0 + 8 * hi) * N + n0 + lm;
        #pragma unroll
        for (int r = 0; r < 8; ++r) {
            crow[(size_t)r * N]      = acc0[r];
            crow[(size_t)r * N + 16] = acc1[r];
        }
    }
}

// -----------------------------------------------------------------------------
// Depthwise causal conv1d (k=4) over the xBC slice of zxbcdt, + bias + SiLU.
// -----------------------------------------------------------------------------
__global__ __launch_bounds__(256) void conv_silu_kernel(
    const float* __restrict__ zx, const float* __restrict__ convw,
    const float* __restrict__ convb, float* __restrict__ xc)
{
    int idx = blockIdx.x * blockDim.x + threadIdx.x;
    if (idx >= M_ROWS * CONV_DIM) return;
    int ch = idx % CONV_DIM;
    int m  = idx / CONV_DIM;
    int b  = m / SEQ;
    int l  = m % SEQ;

    float acc = convb[ch];
    #pragma unroll
    for (int j = 0; j < D_CONV; ++j) {
        int ls = l - (D_CONV - 1) + j;
        if (ls >= 0)
            acc += zx[(size_t)(b * SEQ + ls) * N1 + D_INNER + ch] * convw[ch * D_CONV + j];
    }
    float s = acc / (1.f + expf(-acc));   // SiLU
    xc[(size_t)m * CONV_DIM + ch] = s;
}

// -----------------------------------------------------------------------------
// dt = softplus(dt_raw + dt_bias)
// -----------------------------------------------------------------------------
__global__ __launch_bounds__(256) void dt_kernel(
    const float* __restrict__ zx, const float* __restrict__ dt_bias,
    float* __restrict__ dtb)
{
    int idx = blockIdx.x * blockDim.x + threadIdx.x;
    if (idx >= M_ROWS * NHEADS) return;
    int h = idx % NHEADS;
    int m = idx / NHEADS;
    float v = zx[(size_t)m * N1 + D_INNER + CONV_DIM + h] + dt_bias[h];
    dtb[idx] = (v > 20.f) ? v : log1pf(expf(v));
}

// -----------------------------------------------------------------------------
// SSD chunked scan. One workgroup (256 thr) per (b,h). 64x128 state in LDS.
// Per chunk:  Y = [tril(C B^T) * exp(Acs_l - Acs_s)] (x*dt)  +  exp(Acs) C.state
//             + D*x ;  then state = exp(sumA)*state + sum_l exp(sumA-Acs_l) dt_l x_l B_l^T
// -----------------------------------------------------------------------------
#define NPAD 132   // 128 padded; 132%64=4 -> no stride bank conflicts

__global__ __launch_bounds__(256) void ssd_kernel(
    const float* __restrict__ xc, const float* __restrict__ dtb,
    const float* __restrict__ A_log, const float* __restrict__ Dp,
    float* __restrict__ yss)
{
    __shared__ float sB[CHUNK][NPAD];
    __shared__ float sC[CHUNK][NPAD];
    __shared__ float sx[CHUNK][HEADDIM];
    __shared__ float sdt[CHUNK];
    __shared__ float sG[CHUNK * CHUNK];
    __shared__ float sState[HEADDIM][NPAD];   // [p][n]

    const int tid = threadIdx.x;
    const int bh  = blockIdx.x;
    const int b   = bh >> 5;
    const int h   = bh & 31;

    const float Ah = -expf(A_log[h]);
    const float Dh = Dp[h];

    for (int e = tid; e < HEADDIM * D_STATE; e += 256)
        sState[e >> 7][e & 127] = 0.f;
    __syncthreads();

    for (int c = 0; c < NCHUNK; ++c) {
        const int lbase = c * CHUNK;

        // ---- stage chunk inputs ----
        for (int e = tid; e < CHUNK * D_STATE; e += 256) {
            int l = e >> 7, n = e & 127;
            size_t row = (size_t)(b * SEQ + lbase + l) * CONV_DIM;
            sB[l][n] = xc[row + D_INNER + n];
            sC[l][n] = xc[row + D_INNER + D_STATE + n];
        }
        for (int e = tid; e < CHUNK * HEADDIM; e += 256) {
            int l = e >> 6, p = e & 63;
            sx[l][p] = xc[(size_t)(b * SEQ + lbase + l) * CONV_DIM + h * HEADDIM + p];
        }
        if (tid < CHUNK)
            sdt[tid] = dtb[(size_t)(b * SEQ + lbase + tid) * NHEADS + h];
        __syncthreads();

        // ---- redundant per-thread cumsum of A*dt ----
        float Acs[CHUNK];
        {
            float run = 0.f;
            #pragma unroll
            for (int l = 0; l < CHUNK; ++l) { run += Ah * sdt[l]; Acs[l] = run; }
        }
        const float asum = Acs[CHUNK - 1];

        // ---- G[l][s] = C_l . B_s  (lower triangle) ----
        if (tid < CHUNK * CHUNK) {
            int l = tid / CHUNK, s = tid % CHUNK;
            float g = 0.f;
            if (s <= l)
                for (int n = 0; n < D_STATE; ++n) g += sC[l][n] * sB[s][n];
            sG[tid] = g;
        }
        __syncthreads();

        // ---- Y (reads pre-update state) ----
        for (int e = tid; e < CHUNK * HEADDIM; e += 256) {
            int l = e >> 6, p = e & 63;
            float cs = 0.f;
            for (int n = 0; n < D_STATE; ++n) cs += sC[l][n] * sState[p][n];
            float y = expf(Acs[l]) * cs + Dh * sx[l][p];
            for (int s = 0; s <= l; ++s)
                y += sG[l * CHUNK + s] * expf(Acs[l] - Acs[s]) * sx[s][p] * sdt[s];
            yss[((size_t)(b * SEQ + lbase + l) * NHEADS + h) * HEADDIM + p] = y;
        }
        __syncthreads();

        // ---- state update ----
        const float ea = expf(asum);
        float wdec[CHUNK];
        #pragma unroll
        for (int l = 0; l < CHUNK; ++l) wdec[l] = expf(asum - Acs[l]) * sdt[l];

        for (int e = tid; e < HEADDIM * D_STATE; e += 256) {
            int p = e >> 7, n = e & 127;
            float st = sState[p][n] * ea;
            #pragma unroll
            for (int l = 0; l < CHUNK; ++l)
                st += wdec[l] * sx[l][p] * sB[l][n];
            sState[p][n] = st;
        }
        __syncthreads();
    }
}

// -----------------------------------------------------------------------------
// y = y * silu(z);  y = y * rsqrt(mean(y^2)+1e-5) * norm_w.  One block per row.
// -----------------------------------------------------------------------------
__global__ __launch_bounds__(256) void gate_rms_kernel(
    const float* __restrict__ zx, const float* __restrict__ yss,
    const float* __restrict__ norm_w, float* __restrict__ yn)
{
    const int m   = blockIdx.x;
    const int tid = threadIdx.x;
    const float* zrow = zx  + (size_t)m * N1;
    const float* yrow = yss + (size_t)m * D_INNER;

    float g[8];
    float ss = 0.f;
    #pragma unroll
    for (int j = 0; j < 8; ++j) {
        int i   = tid + j * 256;
        float z = zrow[i];
        float y = yrow[i] * (z / (1.f + expf(-z)));
        g[j] = y;
        ss  += y * y;
    }

    __shared__ float red[256];
    red[tid] = ss;
    __syncthreads();
    for (int off = 128; off > 0; off >>= 1) {
        if (tid < off) red[tid] += red[tid + off];
        __syncthreads();
    }
    const float scale = rsqrtf(red[0] / (float)D_INNER + 1e-5f);

    float* orow = yn + (size_t)m * D_INNER;
    #pragma unroll
    for (int j = 0; j < 8; ++j) {
        int i = tid + j * 256;
        orow[i] = g[j] * scale * norm_w[i];
    }
}

// -----------------------------------------------------------------------------
// Launch pipeline
// -----------------------------------------------------------------------------
extern "C" void kernel_launch(void* const* d_in, const int* in_sizes, int n_in,
                              void* d_out, int out_size, void* d_ws, size_t ws_size,
                              hipStream_t stream)
{
    const float* x       = (const float*)d_in[0];
    const float* W_in    = (const float*)d_in[1];
    const float* conv_w  = (const float*)d_in[2];
    const float* conv_b  = (const float*)d_in[3];
    const float* dt_bias = (const float*)d_in[4];
    const float* A_log   = (const float*)d_in[5];
    const float* Dp      = (const float*)d_in[6];
    const float* norm_w  = (const float*)d_in[7];
    const float* W_out   = (const float*)d_in[8];
    float* out = (float*)d_out;

    // workspace layout (fp32): ~86.5 MB total
    float* zx  = (float*)d_ws;                        // M_ROWS * N1
    float* xc  = zx  + (size_t)M_ROWS * N1;           // M_ROWS * CONV_DIM
    float* dtb = xc  + (size_t)M_ROWS * CONV_DIM;     // M_ROWS * NHEADS
    float* yss = dtb + (size_t)M_ROWS * NHEADS;       // M_ROWS * D_INNER
    float* yn  = yss + (size_t)M_ROWS * D_INNER;      // M_ROWS * D_INNER

    const int Mtiles = M_ROWS / 16;                   // 125, exact

    // 1) zxbcdt = x @ W_in.T        (2000x1024)x(1024x4384)
    {
        dim3 grid((Mtiles + 3) / 4, N1 / 32);
        gemm_f32_wmma<<<grid, 128, LDS_BYTES, stream>>>(x, W_in, zx, Mtiles, D_MODEL, N1);
    }
    // 2) depthwise conv + SiLU
    {
        int total = M_ROWS * CONV_DIM;
        conv_silu_kernel<<<(total + 255) / 256, 256, 0, stream>>>(zx, conv_w, conv_b, xc);
    }
    // 3) dt = softplus(dt + bias)
    {
        int total = M_ROWS * NHEADS;
        dt_kernel<<<(total + 255) / 256, 256, 0, stream>>>(zx, dt_bias, dtb);
    }
    // 4) SSD chunked scan
    ssd_kernel<<<BATCH * NHEADS, 256, 0, stream>>>(xc, dtb, A_log, Dp, yss);
    // 5) gate + RMSNorm
    gate_rms_kernel<<<M_ROWS, 256, 0, stream>>>(zx, yss, norm_w, yn);
    // 6) out = yn @ W_out.T          (2000x2048)x(2048x1024)
    {
        dim3 grid((Mtiles + 3) / 4, D_MODEL / 32);
        gemm_f32_wmma<<<grid, 128, LDS_BYTES, stream>>>(yn, W_out, out, Mtiles, D_INNER, D_MODEL);
    }
}